// TransformerModel_65249143160918
// MI455X (gfx1250) — compile-verified
//
#include <hip/hip_runtime.h>
#include <hip/hip_bf16.h>

// Reference DCE: encoder output is unused; out = (cur @ dec_emb_W.T + dec_emb_b) @ dec_W.T + dec_b
// Fused:  out = cur @ W_c.T + bias_c,  W_c = dec_W @ dec_emb_W,  bias_c = dec_W @ dec_emb_b + dec_b
//
// Bandwidth-bound: 4 MB HBM traffic, ~8.4 MFLOP. We still route the 8x8 apply
// through V_WMMA_F32_16X16X4_F32 (2 WMMAs per 16-row tile, K=8 split into 4+4),
// which is the native CDNA5 f32 matrix path.

typedef __attribute__((ext_vector_type(2))) float v2f;
typedef __attribute__((ext_vector_type(8))) float v8f;

#define D_IN 8
#define D_OUT 8
#define ROWS (256 * 256)       // B * S
#define TILE_ROWS 16
#define N_TILES (ROWS / TILE_ROWS)

// ws layout (floats): [0..63] W_c row-major [o][d'], [64..71] bias_c
__global__ void fuse_weights_kernel(const float* __restrict__ dec_emb_W,
                                    const float* __restrict__ dec_emb_b,
                                    const float* __restrict__ dec_W,
                                    const float* __restrict__ dec_b,
                                    float* __restrict__ wc) {
    int t = threadIdx.x;
    if (t < 64) {
        int o = t >> 3, dp = t & 7;
        float s = 0.f;
#pragma unroll
        for (int d = 0; d < D_IN; ++d)
            s += dec_W[o * D_IN + d] * dec_emb_W[d * D_IN + dp];
        wc[t] = s;
    }
    if (t < 8) {
        float s = dec_b[t];
#pragma unroll
        for (int d = 0; d < D_IN; ++d)
            s += dec_W[t * D_IN + d] * dec_emb_b[d];
        wc[64 + t] = s;
    }
}

__global__ __launch_bounds__(256) void decode_wmma_kernel(
        const float* __restrict__ cur,      // (ROWS, 8) contiguous
        const float* __restrict__ wc,       // fused weights+bias in ws
        float* __restrict__ out,            // (ROWS, 8)
        int n_tiles) {
    const int lane   = threadIdx.x & 31;
    const int wave   = blockIdx.x * (blockDim.x >> 5) + (threadIdx.x >> 5);
    const int nwaves = gridDim.x * (blockDim.x >> 5);

    const int n     = lane & 15;            // N column (output channel if < 8)
    const int khalf = (lane >> 4) << 1;     // K sub-offset: lanes 0-15 -> {k, k+1}, 16-31 -> {k+2, k+3}

    // B operands (4x16 each), columns >= 8 are zero. B[k][n] = W_c[n][k].
    v2f b0 = {0.f, 0.f};   // K chunk 0: input channels 0..3
    v2f b1 = {0.f, 0.f};   // K chunk 1: input channels 4..7
    float bias = 0.f;
    if (n < D_OUT) {
        b0.x = wc[n * D_IN + khalf + 0];
        b0.y = wc[n * D_IN + khalf + 1];
        b1.x = wc[n * D_IN + 4 + khalf + 0];
        b1.y = wc[n * D_IN + 4 + khalf + 1];
        bias = wc[64 + n];
    }
    // C initialized with bias (depends only on N) -> WMMA accumulate does the bias add.
    const v8f cbias = {bias, bias, bias, bias, bias, bias, bias, bias};

    const int m = lane & 15;                // A row within tile

    for (int t = wave; t < n_tiles; t += nwaves) {
        const float* tile = cur + (size_t)t * (TILE_ROWS * D_IN);

        // A (16x4 f32): lane holds row m, K pair starting at khalf (+4 for chunk 1)
        v2f a0, a1;
        a0.x = tile[m * D_IN + khalf + 0];
        a0.y = tile[m * D_IN + khalf + 1];
        a1.x = tile[m * D_IN + 4 + khalf + 0];
        a1.y = tile[m * D_IN + 4 + khalf + 1];

        v8f acc = cbias;
        acc = __builtin_amdgcn_wmma_f32_16x16x4_f32(
            /*neg_a=*/false, a0, /*neg_b=*/false, b0,
            /*c_mod=*/(short)0, acc, /*reuse_a=*/false, /*reuse_b=*/false);
        acc = __builtin_amdgcn_wmma_f32_16x16x4_f32(
            /*neg_a=*/false, a1, /*neg_b=*/false, b1,
            /*c_mod=*/(short)0, acc, /*reuse_a=*/false, /*reuse_b=*/false);

        // D layout: lane N = lane&15, M = r + (lane>=16 ? 8 : 0); keep N < 8 only.
        float* otile = out + (size_t)t * (TILE_ROWS * D_OUT);
        if (n < D_OUT) {
            const int rbase = (lane >> 4) << 3;   // 0 or 8
#pragma unroll
            for (int r = 0; r < 8; ++r) {
                otile[(rbase + r) * D_OUT + n] = acc[r];
            }
        }
    }
}

extern "C" void kernel_launch(void* const* d_in, const int* in_sizes, int n_in,
                              void* d_out, int out_size, void* d_ws, size_t ws_size,
                              hipStream_t stream) {
    // setup_inputs() order:
    //  0 prev_window, 1 current_window, 2 enc_emb_W, 3 enc_emb_b,
    //  4 in_proj_W, 5 in_proj_b, 6 out_proj_W, 7 out_proj_b,
    //  8 lin1_W, 9 lin1_b, 10 lin2_W, 11 lin2_b,
    // 12 ln1_g, 13 ln1_b, 14 ln2_g, 15 ln2_b,
    // 16 dec_emb_W, 17 dec_emb_b, 18 dec_W, 19 dec_b
    const float* cur       = (const float*)d_in[1];
    const float* dec_emb_W = (const float*)d_in[16];
    const float* dec_emb_b = (const float*)d_in[17];
    const float* dec_W     = (const float*)d_in[18];
    const float* dec_b     = (const float*)d_in[19];

    float* wc  = (float*)d_ws;      // 72 floats used
    float* out = (float*)d_out;     // (B, S, OUT) f32

    fuse_weights_kernel<<<1, 64, 0, stream>>>(dec_emb_W, dec_emb_b, dec_W, dec_b, wc);

    // 64 blocks x 8 waves = 512 waves; 4096 tiles -> 8 tiles/wave grid-stride.
    decode_wmma_kernel<<<64, 256, 0, stream>>>(cur, wc, out, N_TILES);
}